// QueryEncDec_80118319939752
// MI455X (gfx1250) — compile-verified
//
#include <hip/hip_runtime.h>

// 128-layer stacked GRU, H=1, T=256 — pure latency problem (0.65 MFLOP, 8 KB).
// 4 waves (wave32), wave w owns layers [32w, 32w+32) as a 32-deep systolic
// pipeline (lane l = layer 32w+l, hidden values hop lanes via __shfl_up).
// Waves are pipelined against each other at CHUNK-timestep granularity through
// LDS boundary buffers + volatile LDS progress flags (per-wave DS ordering on
// CDNA5 makes data-store->flag-store / flag-load->data-load safe without
// fences). Weights/biases/x staged by CDNA5 async DMA
// (global_load_async_to_lds_b128) and one s_wait_asynccnt.
// Makespan: ~287 + 3*(CHUNK+31) ~= 404 sequential cell-steps vs the
// theoretical wavefront floor of 383 (and 1148 for a single-wave version).

#define T_LEN      256
#define LANES      32
#define NWAVES     4
#define CHUNK      8
#define CHUNK_LOG2 3

// LDS layout (float indices)
#define OFF_WIH    0      // 384
#define OFF_WHH    384    // 384
#define OFF_BIH    768    // 384
#define OFF_BHH    1152   // 384
#define OFF_X      1536   // 256 : staged input sequence (feeds wave 0)
#define OFF_B      1792   // 3*256 : boundary buffers wave w -> wave w+1
#define OFF_PROG   2560   // 3 ints : chunks completed by wave w (w=0..2)
#define SMEM_FLOATS 2576

__device__ __forceinline__ void async_b128(unsigned lds_byte_off, const float* g) {
  // GLOBAL_LOAD_ASYNC_TO_LDS_B128 (GV mode): copies 16 bytes per lane from the global address in vaddr to the LDS address in vdst.
  asm volatile("global_load_async_to_lds_b128 %0, %1, off"
               :: "v"(lds_byte_off), "v"((unsigned long long)g)
               : "memory");
}

__device__ __forceinline__ float fast_sigmoid(float x) {
  return __builtin_amdgcn_rcpf(1.0f + __expf(-x));   // v_exp_f32 + v_rcp_f32
}
__device__ __forceinline__ float fast_tanh(float x) {
  return fmaf(2.0f, fast_sigmoid(2.0f * x), -1.0f);  // tanh(x) = 2*sigmoid(2x)-1
}

__global__ __launch_bounds__(NWAVES * LANES)
void gru128_wavepipe(const float* __restrict__ x,
                     const float* __restrict__ w_ih,
                     const float* __restrict__ w_hh,
                     const float* __restrict__ b_ih,
                     const float* __restrict__ b_hh,
                     float* __restrict__ out)
{
  __shared__ float smem[SMEM_FLOATS];
  const int tid  = threadIdx.x;
  const int wave = tid >> 5;      // 0..3  -> layer block
  const int lane = tid & 31;      // 0..31 -> layer within block
  volatile int* prog = (volatile int*)&smem[OFF_PROG];

  // ---- Stage all parameters + input into LDS via CDNA5 async DMA ----
  // Low 32 bits of a flat LDS pointer == wave-relative LDS byte address.
  const unsigned base = (unsigned)(size_t)&smem[0];
  if (tid < 96) {                       // 384 floats = 96 lanes x 16B each
    const unsigned lb = base + (unsigned)(tid * 16);
    async_b128(lb + OFF_WIH * 4u, w_ih + tid * 4);
    async_b128(lb + OFF_WHH * 4u, w_hh + tid * 4);
    async_b128(lb + OFF_BIH * 4u, b_ih + tid * 4);
    async_b128(lb + OFF_BHH * 4u, b_hh + tid * 4);
  }
  if (tid < 64)                         // 256 floats = 64 lanes x 16B
    async_b128(base + OFF_X * 4u + (unsigned)(tid * 16), x + tid * 4);
  if (tid == 0) { prog[0] = 0; prog[1] = 0; prog[2] = 0; }

  asm volatile("s_wait_asynccnt 0" ::: "memory");
  __syncthreads();   // only workgroup barrier; afterwards waves run decoupled

  // ---- Per-lane layer parameters (layer == tid) ----
  const int layer = tid;
  const float wir = smem[OFF_WIH + layer * 3 + 0];
  const float wiz = smem[OFF_WIH + layer * 3 + 1];
  const float win = smem[OFF_WIH + layer * 3 + 2];
  const float whr = smem[OFF_WHH + layer * 3 + 0];
  const float whz = smem[OFF_WHH + layer * 3 + 1];
  const float whn = smem[OFF_WHH + layer * 3 + 2];
  const float bir = smem[OFF_BIH + layer * 3 + 0];
  const float biz = smem[OFF_BIH + layer * 3 + 1];
  const float bin = smem[OFF_BIH + layer * 3 + 2];
  const float bhr = smem[OFF_BHH + layer * 3 + 0];
  const float bhz = smem[OFF_BHH + layer * 3 + 1];
  const float bhn = smem[OFF_BHH + layer * 3 + 2];

  const float* seq_in  = (wave == 0) ? &smem[OFF_X]
                                     : &smem[OFF_B + (wave - 1) * T_LEN];
  float*       seq_out = &smem[OFF_B + wave * T_LEN];     // unused for wave 3
  const bool   last_wave = (wave == NWAVES - 1);

  float h = 0.0f;      // hidden state of this lane's layer
  float carry = 0.0f;  // value forwarded to lane+1 next step

  for (int s = 0; s < T_LEN + LANES - 1; ++s) {
    // Chunk-granularity producer/consumer handshake (wave-uniform branch).
    if (wave != 0 && s < T_LEN && (s & (CHUNK - 1)) == 0) {
      const int need = (s >> CHUNK_LOG2) + 1;
      while (prog[wave - 1] < need)
        __builtin_amdgcn_s_sleep(1);
      asm volatile("" ::: "memory");   // seq_in loads stay below the spin
    }

    // lane l receives lane (l-1)'s output from the previous step
    float xup = __shfl_up(carry, 1, LANES);
    float xin = (lane == 0) ? ((s < T_LEN) ? seq_in[s] : 0.0f) : xup;

    const int t = s - lane;
    const bool active = (t >= 0) && (t < T_LEN);

    // PyTorch GRU cell, gate order [r, z, n], H=1 (all scalar):
    float gr = fmaf(wir, xin, bir) + fmaf(whr, h, bhr);
    float gz = fmaf(wiz, xin, biz) + fmaf(whz, h, bhz);
    float r  = fast_sigmoid(gr);
    float z  = fast_sigmoid(gz);
    float gn = fmaf(win, xin, bin) + r * fmaf(whn, h, bhn);
    float n  = fast_tanh(gn);
    float hn = fmaf(z, h - n, n);          // (1-z)*n + z*h

    if (active) {
      h = hn;
      carry = hn;
      if (lane == LANES - 1) {             // top layer of this wave's block
        if (last_wave) {
          out[t] = hn;                     // layer 127 -> final output
        } else {
          seq_out[t] = hn;                 // feed next wave's block
          if ((t & (CHUNK - 1)) == (CHUNK - 1)) {
            asm volatile("" ::: "memory"); // flag store stays after data store
            prog[wave] = (t + 1) >> CHUNK_LOG2;  // DS in-order per wave
          }
        }
      }
    }
  }
}

extern "C" void kernel_launch(void* const* d_in, const int* in_sizes, int n_in,
                              void* d_out, int out_size, void* d_ws, size_t ws_size,
                              hipStream_t stream) {
  const float* x    = (const float*)d_in[0];
  const float* w_ih = (const float*)d_in[1];
  const float* w_hh = (const float*)d_in[2];
  const float* b_ih = (const float*)d_in[3];
  const float* b_hh = (const float*)d_in[4];
  float* out = (float*)d_out;
  gru128_wavepipe<<<1, NWAVES * LANES, 0, stream>>>(x, w_ih, w_hh, b_ih, b_hh, out);
}